// SSD_34892314313729
// MI455X (gfx1250) — compile-verified
//
#include <hip/hip_runtime.h>
#include <cstdint>

#define NC      21     // classes incl. background
#define NCLS    20     // foreground classes
#define TOPK    200
#define CONF_T  0.01f
#define NMS_T   0.45f
#define BB      64
#define PP      8732
#define CAND    (NC * TOPK)   // 4200
#define NT      256           // threads per block (8 wave32)

// stable "better" comparator: larger value wins; ties -> lower index
// (matches jax.lax.top_k / stable argsort(-score) semantics)
__device__ __forceinline__ void take_better(float& bv, int& bi, float v, int i) {
    if (v > bv || (v == bv && i < bi)) { bv = v; bi = i; }
}

__device__ __forceinline__ void wave_argmax(float& bv, int& bi) {
    // wave32 butterfly reduction
    #pragma unroll
    for (int off = 16; off > 0; off >>= 1) {
        float ov = __shfl_xor(bv, off, 32);
        int   oi = __shfl_xor(bi, off, 32);
        take_better(bv, bi, ov, oi);
    }
}

// 256-thread (8-wave) block argmax; result valid in thread 0 only.
// Contains one __syncthreads.
__device__ __forceinline__ void block_argmax(float& bv, int& bi,
                                             float* rv, int* ri, int tid) {
    wave_argmax(bv, bi);
    const int wid = tid >> 5;
    if ((tid & 31) == 0) { rv[wid] = bv; ri[wid] = bi; }
    __syncthreads();
    if (tid == 0) {
        #pragma unroll
        for (int w = 1; w < 8; ++w) take_better(bv, bi, rv[w], ri[w]);
    }
}

// ---------------------------------------------------------------------------
// Kernel A: softmax over 21 classes + box decode. One thread per (b,p).
// ---------------------------------------------------------------------------
__global__ void __launch_bounds__(NT)
ssd_softmax_decode(const float* __restrict__ loc,
                   const float* __restrict__ conf,
                   const float* __restrict__ priors,
                   float* __restrict__ decoded,   // [B][P][4]
                   float* __restrict__ scores)    // [B][20][P] (class-major)
{
    const long long t = (long long)blockIdx.x * NT + threadIdx.x;
    if (t >= (long long)BB * PP) return;
    const int b = (int)(t / PP);
    const int p = (int)(t - (long long)b * PP);

    float x[NC];
    const float* cf = conf + (size_t)t * NC;
    #pragma unroll
    for (int i = 0; i < NC; ++i) x[i] = cf[i];
    float m = x[0];
    #pragma unroll
    for (int i = 1; i < NC; ++i) m = fmaxf(m, x[i]);
    float s = 0.f;
    #pragma unroll
    for (int i = 0; i < NC; ++i) { x[i] = __expf(x[i] - m); s += x[i]; }
    const float inv = 1.f / s;
    #pragma unroll
    for (int i = 1; i < NC; ++i)
        scores[((size_t)b * NCLS + (i - 1)) * PP + p] = x[i] * inv;

    const float4 l  = ((const float4*)loc)[t];
    const float4 pr = ((const float4*)priors)[p];
    const float cx = pr.x + l.x * 0.1f * pr.z;
    const float cy = pr.y + l.y * 0.1f * pr.w;
    const float w  = pr.z * __expf(l.z * 0.2f);
    const float h  = pr.w * __expf(l.w * 0.2f);
    const float x1 = cx - 0.5f * w;
    const float y1 = cy - 0.5f * h;
    ((float4*)decoded)[t] = make_float4(x1, y1, x1 + w, y1 + h);
}

// ---------------------------------------------------------------------------
// Kernel B: per (image, class): top-200 + greedy NMS. One block per (b,c).
// Scores staged into LDS via gfx1250 async global->LDS copy; incremental
// tournament argmax: per-iteration cost ~256-entry reduce + one 35-elem
// segment rescan instead of a full 8732 sweep.
// ---------------------------------------------------------------------------
__global__ void __launch_bounds__(NT)
ssd_class_topk_nms(const float* __restrict__ scores,   // [B][20][P]
                   const float* __restrict__ decoded,  // [B][P][4]
                   float* __restrict__ rows)           // [B][21][200][5]
{
    __shared__ __align__(16) float s_sc[PP];           // 34,928 B
    __shared__ float lbv[NT];                          // per-thread segment best
    __shared__ int   lbi[NT];
    __shared__ float rv[8];
    __shared__ int   ri[8];
    __shared__ float sel_s[TOPK];
    __shared__ int   sel_i[TOPK];
    __shared__ float bx1[TOPK], by1[TOPK], bx2[TOPK], by2[TOPK], bar[TOPK];
    __shared__ unsigned char sup[TOPK];
    __shared__ int   pos[TOPK];
    __shared__ int   s_nv;
    __shared__ int   s_rem;

    const int tid = threadIdx.x;
    const int bc  = blockIdx.x;
    const int b   = bc / NCLS;
    const int c   = bc % NCLS;

    const float* src = scores + (size_t)bc * PP;

    // ---- async bulk copy: global -> LDS, 16B per lane-op (ASYNCcnt path) ----
    const unsigned lds0 = (unsigned)(unsigned long long)(uintptr_t)&s_sc[0];
    for (int i = tid; i < PP / 4; i += NT) {
        const float* g = src + (size_t)i * 4;
        const unsigned l = lds0 + (unsigned)i * 16u;
        asm volatile("global_load_async_to_lds_b128 %0, %1, off"
                     :: "v"(l), "v"(g) : "memory");
    }
    asm volatile("s_wait_asynccnt 0x0" ::: "memory");
    if (tid == 0) s_nv = TOPK;
    __syncthreads();

    // threshold transform (reference: s = score>CONF ? score : -1) fused
    // with initial per-segment argmax (segment = indices j with j%256==tid)
    {
        float v0 = -1e30f; int i0 = 0x7fffffff;
        for (int j = tid; j < PP; j += NT) {
            float v = s_sc[j];
            v = (v > CONF_T) ? v : -1.0f;
            s_sc[j] = v;
            take_better(v0, i0, v, j);
        }
        lbv[tid] = v0; lbi[tid] = i0;
    }
    __syncthreads();

    // ---- iterative stable top-200 via incremental tournament ----
    for (int k = 0; k < TOPK; ++k) {
        if (k >= s_nv) break;                 // uniform across block
        float bv = lbv[tid]; int bi = lbi[tid];
        block_argmax(bv, bi, rv, ri, tid);    // one __syncthreads inside
        if (tid == 0) {
            if (bv <= CONF_T) {
                s_nv = k;                     // remaining slots invalid
            } else {
                sel_s[k] = bv; sel_i[k] = bi;
                s_sc[bi] = -1e30f;            // remove winner
                s_rem = bi;
            }
        }
        __syncthreads();
        if (k >= s_nv) break;                 // uniform: selection finished
        // only the owner of the removed element rescans its segment
        if ((s_rem & (NT - 1)) == tid) {
            float v0 = -1e30f; int i0 = 0x7fffffff;
            for (int j = tid; j < PP; j += NT) take_better(v0, i0, s_sc[j], j);
            lbv[tid] = v0; lbi[tid] = i0;
        }
        __syncthreads();
    }
    const int nv = s_nv;

    // ---- gather decoded boxes + areas ----
    const float4* dbox = (const float4*)(decoded + (size_t)b * PP * 4);
    for (int t = tid; t < nv; t += NT) {
        const float4 q = dbox[sel_i[t]];
        bx1[t] = q.x; by1[t] = q.y; bx2[t] = q.z; by2[t] = q.w;
        bar[t] = (q.z - q.x + 1.f) * (q.w - q.y + 1.f);
        sup[t] = 0;
    }
    __syncthreads();

    // ---- sequential greedy NMS (matches reference fori_loop) ----
    for (int i = 0; i < nv; ++i) {
        if (!sup[i]) {                        // uniform (shared read)
            const float xi1 = bx1[i], yi1 = by1[i];
            const float xi2 = bx2[i], yi2 = by2[i], ai = bar[i];
            for (int t = tid; t < nv; t += NT) {
                if (t > i && !sup[t]) {
                    const float xx1 = fmaxf(xi1, bx1[t]);
                    const float yy1 = fmaxf(yi1, by1[t]);
                    const float xx2 = fminf(xi2, bx2[t]);
                    const float yy2 = fminf(yi2, by2[t]);
                    const float w = fmaxf(0.f, xx2 - xx1 + 1.f);
                    const float h = fmaxf(0.f, yy2 - yy1 + 1.f);
                    const float inter = w * h;
                    const float iou = inter / (ai + bar[t] - inter);
                    if (iou > NMS_T) sup[t] = 1;
                }
            }
        }
        __syncthreads();
    }

    // ---- stable compaction: kept rows first, in slot (=score) order ----
    if (tid == 0) {
        int cnt = 0;
        for (int t = 0; t < nv; ++t) pos[t] = sup[t] ? -1 : cnt++;
    }
    __syncthreads();

    float* orow = rows + ((size_t)b * NC + (c + 1)) * TOPK * 5;
    for (int j = tid; j < TOPK * 5; j += NT) orow[j] = 0.f;
    if (c == 0) {   // background class rows of this image
        float* bg = rows + (size_t)b * NC * TOPK * 5;
        for (int j = tid; j < TOPK * 5; j += NT) bg[j] = 0.f;
    }
    __syncthreads();
    for (int t = tid; t < nv; t += NT) {
        const int r = pos[t];
        if (r >= 0) {
            orow[r * 5 + 0] = sel_s[t];
            orow[r * 5 + 1] = bx1[t];
            orow[r * 5 + 2] = by1[t];
            orow[r * 5 + 3] = bx2[t];
            orow[r * 5 + 4] = by2[t];
        }
    }
}

// ---------------------------------------------------------------------------
// Kernel C: per image final top-200 over 4200 candidates, emitted in
// ascending original index (stable mask-compaction of the reference).
// Incremental tournament argmax, segment size ceil(4200/256)=17.
// ---------------------------------------------------------------------------
__global__ void __launch_bounds__(NT)
ssd_final_select(const float* __restrict__ rows,  // [B][21][200][5]
                 float* __restrict__ out)         // [B*200][7]
{
    __shared__ float sc[CAND];
    __shared__ unsigned char msk[CAND];
    __shared__ float lbv[NT];
    __shared__ int   lbi[NT];
    __shared__ float rv[8];
    __shared__ int   ri[8];
    __shared__ int   ord[TOPK];
    __shared__ int   s_rem;

    const int tid = threadIdx.x;
    const int b   = blockIdx.x;
    const float* rimg = rows + (size_t)b * NC * TOPK * 5;

    // load scores (stride 5) + initial per-segment argmax
    {
        float v0 = -1e30f; int i0 = 0x7fffffff;
        for (int j = tid; j < CAND; j += NT) {
            const float v = rimg[(size_t)j * 5];
            sc[j] = v; msk[j] = 0;
            take_better(v0, i0, v, j);
        }
        lbv[tid] = v0; lbi[tid] = i0;
    }
    __syncthreads();

    for (int k = 0; k < TOPK; ++k) {
        float bv = lbv[tid]; int bi = lbi[tid];
        block_argmax(bv, bi, rv, ri, tid);    // one __syncthreads inside
        if (tid == 0) { msk[bi] = 1; sc[bi] = -1e30f; s_rem = bi; }
        __syncthreads();
        if ((s_rem & (NT - 1)) == tid) {
            float v0 = -1e30f; int i0 = 0x7fffffff;
            for (int j = tid; j < CAND; j += NT) take_better(v0, i0, sc[j], j);
            lbv[tid] = v0; lbi[tid] = i0;
        }
        __syncthreads();
    }

    if (tid == 0) {
        int p = 0;
        for (int j = 0; j < CAND; ++j) if (msk[j]) ord[p++] = j;
    }
    __syncthreads();

    if (tid < TOPK) {
        const int j = ord[tid];
        const float* r = rimg + (size_t)j * 5;
        float* o = out + ((size_t)b * TOPK + tid) * 7;
        o[0] = r[0]; o[1] = r[1]; o[2] = r[2];
        o[3] = r[3]; o[4] = r[4]; o[5] = 0.f; o[6] = 0.f;
    }
}

// ---------------------------------------------------------------------------
extern "C" void kernel_launch(void* const* d_in, const int* in_sizes, int n_in,
                              void* d_out, int out_size, void* d_ws, size_t ws_size,
                              hipStream_t stream) {
    const float* loc    = (const float*)d_in[0];
    const float* conf   = (const float*)d_in[1];
    const float* priors = (const float*)d_in[2];
    float* out = (float*)d_out;

    // workspace layout (floats): decoded | scores^T | rows  (~59 MB total)
    float* decoded = (float*)d_ws;                       // B*P*4
    float* scores  = decoded + (size_t)BB * PP * 4;      // B*20*P
    float* rows    = scores  + (size_t)BB * NCLS * PP;   // B*21*200*5

    const int totalA = BB * PP;
    ssd_softmax_decode<<<(totalA + NT - 1) / NT, NT, 0, stream>>>(
        loc, conf, priors, decoded, scores);
    ssd_class_topk_nms<<<BB * NCLS, NT, 0, stream>>>(scores, decoded, rows);
    ssd_final_select<<<BB, NT, 0, stream>>>(rows, out);
}